// RegistrationLoss_80556406604138
// MI455X (gfx1250) — compile-verified
//
#include <hip/hip_runtime.h>
#include <cstdint>
#include <cstddef>

#define TPB    256   // threads per block (8 waves of 32)
#define TILE   512   // points staged per tile
#define PPT    2     // points per thread per tile (TILE/TPB)
#define NBLK   1024  // grid size for main kernel

// ---------------------------------------------------------------------------
// CDNA5 async global -> LDS copy (gfx1250). Per-lane: LDS byte address in a
// 32-bit VGPR, 64-bit global address in a VGPR pair. Tracked by ASYNCcnt.
// ---------------------------------------------------------------------------
__device__ __forceinline__ unsigned lds_off_u32(const float* p) {
  // Generic pointer to LDS: addr[31:0] is the wave-relative LDS address
  // (CDNA5 ISA 10.2 aperture mapping), directly usable by async-to-LDS ops.
  return (unsigned)(uintptr_t)p;
}

__device__ __forceinline__ void async_b128(unsigned lds_addr, const float* g) {
  asm volatile("global_load_async_to_lds_b128 %0, %1, off"
               :: "v"(lds_addr), "v"((unsigned long long)(uintptr_t)g)
               : "memory");
}

__device__ __forceinline__ void wait_async_le6() {
  asm volatile("s_wait_asynccnt 0x6" ::: "memory");
}
__device__ __forceinline__ void wait_async_le0() {
  asm volatile("s_wait_asynccnt 0x0" ::: "memory");
}
__device__ __forceinline__ void wait_ds0() {
  asm volatile("s_wait_dscnt 0x0" ::: "memory");
}

// ---------------------------------------------------------------------------
// SE(3) math (fp32, matches reference formulas; K^2 = w w^T - |w|^2 I closed
// form instead of an explicit 3x3 matmul)
// ---------------------------------------------------------------------------
__device__ __forceinline__ void exp_se3(const float* __restrict__ xi,
                                        float* __restrict__ R,
                                        float* __restrict__ t) {
  float x = xi[0], y = xi[1], z = xi[2];
  float a = xi[3], b = xi[4], c = xi[5];
  float t2 = x * x + y * y + z * z;
  bool sm = t2 < 1e-8f;
  float t2s = sm ? 1.0f : t2;
  float th = sqrtf(t2s);
  float sn, cs;
  sincosf(th, &sn, &cs);
  float A = sm ? 1.0f - t2 * (1.0f / 6.0f) : sn / th;
  float B = sm ? 0.5f - t2 * (1.0f / 24.0f) : (1.0f - cs) / t2s;
  float C = sm ? (1.0f / 6.0f) - t2 * (1.0f / 120.0f) : (th - sn) / (t2s * th);
  float xx = x * x, yy = y * y, zz = z * z;
  float xy = x * y, xz = x * z, yz = y * z;
  R[0] = 1.0f + B * (xx - t2); R[1] = -A * z + B * xy;      R[2] =  A * y + B * xz;
  R[3] =  A * z + B * xy;      R[4] = 1.0f + B * (yy - t2); R[5] = -A * x + B * yz;
  R[6] = -A * y + B * xz;      R[7] =  A * x + B * yz;      R[8] = 1.0f + B * (zz - t2);
  float V0 = 1.0f + C * (xx - t2), V1 = -B * z + C * xy,      V2 =  B * y + C * xz;
  float V3 =  B * z + C * xy,      V4 = 1.0f + C * (yy - t2), V5 = -B * x + C * yz;
  float V6 = -B * y + C * xz,      V7 =  B * x + C * yz,      V8 = 1.0f + C * (zz - t2);
  t[0] = V0 * a + V1 * b + V2 * c;
  t[1] = V3 * a + V4 * b + V5 * c;
  t[2] = V6 * a + V7 * b + V8 * c;
}

__device__ __forceinline__ float point_loss(const float* __restrict__ xp,
                                            const float* __restrict__ xt) {
  float Rp[9], tp[3], Rt[9], tt[3];
  exp_se3(xp, Rp, tp);
  exp_se3(xt, Rt, tt);

  // Rd = Rp^T * Rt ; td = Rp^T * (tt - tp)
  float rd[9];
#pragma unroll
  for (int i = 0; i < 3; ++i)
#pragma unroll
    for (int k = 0; k < 3; ++k)
      rd[i * 3 + k] = Rp[0 + i] * Rt[0 + k] + Rp[3 + i] * Rt[3 + k] + Rp[6 + i] * Rt[6 + k];
  float d0 = tt[0] - tp[0], d1 = tt[1] - tp[1], d2 = tt[2] - tp[2];
  float td0 = Rp[0] * d0 + Rp[3] * d1 + Rp[6] * d2;
  float td1 = Rp[1] * d0 + Rp[4] * d1 + Rp[7] * d2;
  float td2 = Rp[2] * d0 + Rp[5] * d1 + Rp[8] * d2;

  // log_se3
  float tr = rd[0] + rd[4] + rd[8];
  float cc = fminf(fmaxf((tr - 1.0f) * 0.5f, -1.0f + 1e-7f), 1.0f - 1e-7f);
  float th = acosf(cc);
  bool sm = th < 1e-4f;
  float s = sinf(th);
  float ths = sm ? 1.0f : th;
  float ss  = sm ? 1.0f : s;
  float coef = sm ? 0.5f + th * th * (1.0f / 12.0f) : th / (2.0f * ss);
  float wx = coef * (rd[7] - rd[5]);
  float wy = coef * (rd[2] - rd[6]);
  float wz = coef * (rd[3] - rd[1]);
  float w2 = wx * wx + wy * wy + wz * wz;
  float omc = sm ? 1.0f : (1.0f - cc);
  float D = sm ? (1.0f / 12.0f)
               : (1.0f - th * s / (2.0f * omc)) / (ths * ths);
  // Vinv = I - 0.5*K + D*(w w^T - w2 I)
  float vi00 = 1.0f + D * (wx * wx - w2);
  float vi01 =  0.5f * wz + D * wx * wy;
  float vi02 = -0.5f * wy + D * wx * wz;
  float vi10 = -0.5f * wz + D * wx * wy;
  float vi11 = 1.0f + D * (wy * wy - w2);
  float vi12 =  0.5f * wx + D * wy * wz;
  float vi20 =  0.5f * wy + D * wx * wz;
  float vi21 = -0.5f * wx + D * wy * wz;
  float vi22 = 1.0f + D * (wz * wz - w2);
  float v0 = vi00 * td0 + vi01 * td1 + vi02 * td2;
  float v1 = vi10 * td0 + vi11 * td1 + vi12 * td2;
  float v2 = vi20 * td0 + vi21 * td1 + vi22 * td2;
  return w2 + v0 * v0 + v1 * v1 + v2 * v2;
}

// ---------------------------------------------------------------------------
// Stage one tile's worth of this thread's 2 points (48 B each array) to LDS
// via 6 async b128 loads. Each thread later reads only what it wrote -> no
// workgroup barrier needed, only s_wait_asynccnt.
// ---------------------------------------------------------------------------
__device__ __forceinline__ void stage_tile(const float* __restrict__ predict,
                                           const float* __restrict__ target,
                                           int tile, unsigned lp,
                                           unsigned lds_p, unsigned lds_t) {
  size_t fidx = ((size_t)tile * TILE + lp) * 6;  // 16B-aligned (lp even)
  const float* gp = predict + fidx;
  const float* gt = target + fidx;
  unsigned lo = lp * 24;                         // byte offset, 16B-aligned
  wait_ds0();  // prior DS reads of this buffer fully drained before overwrite
  async_b128(lds_p + lo + 0,  gp + 0);
  async_b128(lds_p + lo + 16, gp + 4);
  async_b128(lds_p + lo + 32, gp + 8);
  async_b128(lds_t + lo + 0,  gt + 0);
  async_b128(lds_t + lo + 16, gt + 4);
  async_b128(lds_t + lo + 32, gt + 8);
}

__global__ void __launch_bounds__(TPB)
se3_loss_main(const float* __restrict__ predict, const float* __restrict__ target,
              float* __restrict__ partials, int ntiles) {
  __shared__ alignas(16) float lds[2][2][TILE * 6];  // [buf][pred|targ] 48 KB
  __shared__ float red[TPB];

  const unsigned tid = threadIdx.x;
  const unsigned lp = tid * PPT;  // local point index (even)
  unsigned lb[2][2];
#pragma unroll
  for (int bf = 0; bf < 2; ++bf)
#pragma unroll
    for (int ar = 0; ar < 2; ++ar) lb[bf][ar] = lds_off_u32(&lds[bf][ar][0]);

  float acc = 0.0f;
  int tile = (int)blockIdx.x;
  const int stride = (int)gridDim.x;
  if (tile < ntiles) stage_tile(predict, target, tile, lp, lb[0][0], lb[0][1]);
  int buf = 0;
  for (; tile < ntiles; tile += stride) {
    int nxt = tile + stride;
    if (nxt < ntiles) {
      stage_tile(predict, target, nxt, lp, lb[buf ^ 1][0], lb[buf ^ 1][1]);
      wait_async_le6();   // current tile's 6 loads complete; next 6 in flight
    } else {
      wait_async_le0();
    }
    const float* sp = &lds[buf][0][lp * 6];
    const float* st = &lds[buf][1][lp * 6];
    acc += point_loss(sp, st);
    acc += point_loss(sp + 6, st + 6);
    buf ^= 1;
  }

  red[tid] = acc;
  __syncthreads();
#pragma unroll
  for (unsigned off = TPB / 2; off > 0; off >>= 1) {
    if (tid < off) red[tid] += red[tid + off];
    __syncthreads();
  }
  if (tid == 0) partials[blockIdx.x] = red[0];
}

__global__ void __launch_bounds__(256)
se3_loss_finish(const float* __restrict__ partials, int nparts,
                const float* __restrict__ predict, const float* __restrict__ target,
                int tail_start, int n, float* __restrict__ out) {
  __shared__ double sd[256];
  double s = 0.0;
  for (int i = (int)threadIdx.x; i < nparts; i += 256) s += (double)partials[i];
  // generic tail (empty when N % TILE == 0, as here)
  for (int p = tail_start + (int)threadIdx.x; p < n; p += 256) {
    float xp[6], xt[6];
#pragma unroll
    for (int j = 0; j < 6; ++j) {
      xp[j] = predict[(size_t)p * 6 + j];
      xt[j] = target[(size_t)p * 6 + j];
    }
    s += (double)point_loss(xp, xt);
  }
  sd[threadIdx.x] = s;
  __syncthreads();
#pragma unroll
  for (int off = 128; off > 0; off >>= 1) {
    if ((int)threadIdx.x < off) sd[threadIdx.x] += sd[threadIdx.x + off];
    __syncthreads();
  }
  if (threadIdx.x == 0) out[0] = (float)(sd[0] / (double)n);
}

extern "C" void kernel_launch(void* const* d_in, const int* in_sizes, int n_in,
                              void* d_out, int out_size, void* d_ws, size_t ws_size,
                              hipStream_t stream) {
  (void)n_in; (void)out_size; (void)ws_size;
  const float* predict = (const float*)d_in[0];
  const float* target  = (const float*)d_in[1];
  float* out = (float*)d_out;
  float* partials = (float*)d_ws;

  const int n = in_sizes[0] / 6;       // 2,097,152
  const int ntiles = n / TILE;         // 4096
  const int tail_start = ntiles * TILE;

  se3_loss_main<<<dim3(NBLK), dim3(TPB), 0, stream>>>(predict, target, partials, ntiles);
  se3_loss_finish<<<dim3(1), dim3(256), 0, stream>>>(partials, NBLK, predict, target,
                                                     tail_start, n, out);
}